// RoIHeadTemplate_82325933130327
// MI455X (gfx1250) — compile-verified
//
#include <hip/hip_runtime.h>
#include <hip/hip_bf16.h>
#include <math.h>

#define BB 8
#define NN 16384
#define CC 3
#define KPRE 1024
#define POST 512
#define NMS_TH 0.7f
#define EPSF 1e-8f

typedef __attribute__((ext_vector_type(2))) float v2f;
typedef __attribute__((ext_vector_type(8))) float v8f;

// ---------------------------------------------------------------------------
// Kernel 1a: per-box score/key. key = sortable(score)<<32 | (~idx) so that a
// descending sort puts higher scores first and, on ties, lower indices first
// (matches jax.lax.top_k ordering).
// ---------------------------------------------------------------------------
__global__ __launch_bounds__(256) void score_key_kernel(
    const float* __restrict__ cls, unsigned long long* __restrict__ keys) {
  int b = blockIdx.y;
  int i = blockIdx.x * 256 + threadIdx.x;
  const float* c = cls + ((size_t)b * NN + i) * CC;
  float s = c[0];
  if (c[1] > s) s = c[1];
  if (c[2] > s) s = c[2];
  unsigned bits = __float_as_uint(s);
  unsigned sb = (bits & 0x80000000u) ? ~bits : (bits ^ 0x80000000u);
  keys[(size_t)b * NN + i] =
      ((unsigned long long)sb << 32) | (unsigned)(0xFFFFFFFFu - (unsigned)i);
}

// ---------------------------------------------------------------------------
// Kernel 1b: one compare-exchange stage of a global bitonic sort (descending).
// ---------------------------------------------------------------------------
__global__ __launch_bounds__(256) void bitonic_step(
    unsigned long long* __restrict__ keys, int k, int j) {
  int b = blockIdx.y;
  int t = blockIdx.x * 256 + threadIdx.x;
  unsigned long long* kk = keys + (size_t)b * NN;
  int ixj = t ^ j;
  if (ixj > t) {
    unsigned long long a = kk[t], c = kk[ixj];
    bool up = (t & k) == 0;            // up-blocks sorted descending overall
    bool sw = up ? (a < c) : (a > c);
    if (sw) { kk[t] = c; kk[ixj] = a; }
  }
}

// ---------------------------------------------------------------------------
// Kernel 1c: gather top-KPRE boxes + per-box NMS precompute (center, |c|^2,
// circumscribed BEV radius), recover score, recompute argmax label.
// ---------------------------------------------------------------------------
__global__ __launch_bounds__(256) void gather_kernel(
    const float* __restrict__ boxes, const float* __restrict__ cls,
    const unsigned long long* __restrict__ keys,
    float* __restrict__ tb, float* __restrict__ ts, int* __restrict__ tl,
    float* __restrict__ cx, float* __restrict__ cy,
    float* __restrict__ pp, float* __restrict__ rad) {
  int b = blockIdx.y;
  int k = blockIdx.x * 256 + threadIdx.x;   // 0..KPRE-1
  unsigned long long key = keys[(size_t)b * NN + k];
  unsigned idx = 0xFFFFFFFFu - (unsigned)(key & 0xFFFFFFFFull);
  unsigned sb = (unsigned)(key >> 32);
  unsigned bits = (sb & 0x80000000u) ? (sb ^ 0x80000000u) : ~sb;
  float score = __uint_as_float(bits);

  const float* bp = boxes + ((size_t)b * NN + idx) * 7;
  float bx[7];
#pragma unroll
  for (int q = 0; q < 7; ++q) bx[q] = bp[q];
#pragma unroll
  for (int q = 0; q < 7; ++q) tb[((size_t)b * KPRE + k) * 7 + q] = bx[q];

  const float* cp = cls + ((size_t)b * NN + idx) * CC;
  int lab = 0; float m = cp[0];
  if (cp[1] > m) { m = cp[1]; lab = 1; }
  if (cp[2] > m) { m = cp[2]; lab = 2; }

  size_t o = (size_t)b * KPRE + k;
  ts[o] = score;
  tl[o] = lab;
  cx[o] = bx[0];
  cy[o] = bx[1];
  pp[o] = bx[0] * bx[0] + bx[1] * bx[1];
  rad[o] = 0.5f * sqrtf(bx[3] * bx[3] + bx[4] * bx[4]);
}

// ---------------------------------------------------------------------------
// Precomputed rotated box: center, half-dims, sincos, area, BEV corners.
// ---------------------------------------------------------------------------
struct RBox {
  float x, y, hx, hy, cs, sn, area;
  float cx4[4], cy4[4];
};

__device__ __forceinline__ void mk_rbox(const float* b7, RBox& R) {
  float s, c;
  sincosf(b7[6], &s, &c);
  R.x = b7[0]; R.y = b7[1];
  R.hx = 0.5f * b7[3]; R.hy = 0.5f * b7[4];
  R.cs = c; R.sn = s;
  R.area = b7[3] * b7[4];
  const float tx[4] = {1.f, 1.f, -1.f, -1.f};
  const float ty[4] = {1.f, -1.f, -1.f, 1.f};
#pragma unroll
  for (int k = 0; k < 4; ++k) {
    float px = tx[k] * R.hx, py = ty[k] * R.hy;
    R.cx4[k] = c * px - s * py + R.x;
    R.cy4[k] = s * px + c * py + R.y;
  }
}

__device__ __forceinline__ float xprod(float ux, float uy, float vx, float vy) {
  return ux * vy - uy * vx;
}

__device__ __forceinline__ bool pt_in_rbox(float qx, float qy, const RBox& B) {
  float dx = qx - B.x, dy = qy - B.y;
  float lx = B.cs * dx + B.sn * dy;
  float ly = -B.sn * dx + B.cs * dy;
  return (fabsf(lx) <= B.hx + 1e-5f) && (fabsf(ly) <= B.hy + 1e-5f);
}

// Diamond pseudo-angle: strictly monotone in atan2(dy,dx) over (-pi,pi],
// range (-2,2]. Identical argsort order at ~1/20 the VALU cost of atan2f.
// Degenerate (0,0) -> 0 (mid-range), matching atan2(0,0)=0.
__device__ __forceinline__ float pseudo_angle(float dy, float dx) {
  float ax = fabsf(dx), ay = fabsf(dy);
  float den = ax + ay;
  if (den < 1e-30f) return 0.f;
  float p = 1.f - dx / den;              // in [0,2]
  return copysignf(p, dy);               // dy>=0 -> [0,2], dy<0 -> [-2,0)
}

// compare-exchange for the sorting network (register-resident)
__device__ __forceinline__ void ce(float& a, float& b, float& x0, float& x1,
                                   float& y0, float& y1) {
  if (a > b) {
    float t = a; a = b; b = t;
    t = x0; x0 = x1; x1 = t;
    t = y0; y0 = y1; y1 = t;
  }
}

// ---------------------------------------------------------------------------
// Exact rotated-rectangle intersection (candidate-vertex method, matches ref).
// All 24-slot arrays use only compile-time indices (Batcher sorting network),
// so everything stays in VGPRs -- no scratch traffic.
// ---------------------------------------------------------------------------
__device__ float rotated_inter_area(const RBox& A, const RBox& B) {
  float px[24], py[24];
  bool vb[24];
#pragma unroll
  for (int ia = 0; ia < 4; ++ia) {
    int ia2 = (ia + 1) & 3;
    float dax = A.cx4[ia2] - A.cx4[ia], day = A.cy4[ia2] - A.cy4[ia];
#pragma unroll
    for (int ib = 0; ib < 4; ++ib) {
      int ib2 = (ib + 1) & 3;
      float dbx = B.cx4[ib2] - B.cx4[ib], dby = B.cy4[ib2] - B.cy4[ib];
      float dfx = B.cx4[ib] - A.cx4[ia], dfy = B.cy4[ib] - A.cy4[ia];
      float den = xprod(dax, day, dbx, dby);
      float safe = (fabsf(den) < EPSF) ? 1.0f : den;
      float rinv = 1.0f / safe;
      float t = xprod(dfx, dfy, dbx, dby) * rinv;
      float u = xprod(dfx, dfy, dax, day) * rinv;
      bool ok = (fabsf(den) >= EPSF) && (t >= 0.f) && (t <= 1.f) && (u >= 0.f) && (u <= 1.f);
      int id = ia * 4 + ib;
      px[id] = A.cx4[ia] + t * dax;
      py[id] = A.cy4[ia] + t * day;
      vb[id] = ok;
    }
  }
#pragma unroll
  for (int k = 0; k < 4; ++k) {
    px[16 + k] = A.cx4[k]; py[16 + k] = A.cy4[k]; vb[16 + k] = pt_in_rbox(A.cx4[k], A.cy4[k], B);
    px[20 + k] = B.cx4[k]; py[20 + k] = B.cy4[k]; vb[20 + k] = pt_in_rbox(B.cx4[k], B.cy4[k], A);
  }

  int n = 0; float sx = 0.f, sy = 0.f;
#pragma unroll
  for (int i = 0; i < 24; ++i)
    if (vb[i]) { ++n; sx += px[i]; sy += py[i]; }
  float inv = 1.f / fmaxf((float)n, 1.f);
  float cxx = sx * inv, cyy = sy * inv;

  float ang[24];
#pragma unroll
  for (int i = 0; i < 24; ++i)
    ang[i] = vb[i] ? pseudo_angle(py[i] - cyy, px[i] - cxx) : 1e9f;

  // Batcher odd-even mergesort network, n = 24 (invalids -> end, ang = 1e9).
  // Fully unrolled: all indices are compile-time constants.
#pragma unroll
  for (int p = 1; p < 24; p <<= 1) {
#pragma unroll
    for (int k = p; k >= 1; k >>= 1) {
#pragma unroll
      for (int j = k & (p - 1); j + k < 24; j += 2 * k) {
#pragma unroll
        for (int i = 0; i < k; ++i) {
          if (i + j + k < 24) {
            if (((i + j) / (2 * p)) == ((i + j + k) / (2 * p))) {
              ce(ang[i + j], ang[i + j + k], px[i + j], px[i + j + k],
                 py[i + j], py[i + j + k]);
            }
          }
        }
      }
    }
  }

  // dead slots -> first (lowest-angle, valid if any) point; recenter
  float fx = px[0], fy = py[0];
#pragma unroll
  for (int i = 0; i < 24; ++i) {
    bool v = ang[i] < 1e8f;
    px[i] = (v ? px[i] : fx) - cxx;
    py[i] = (v ? py[i] : fy) - cyy;
  }
  float s2 = 0.f;
#pragma unroll
  for (int i = 0; i < 24; ++i) {
    int j = (i + 1) == 24 ? 0 : (i + 1);
    s2 += xprod(px[i], py[i], px[j], py[j]);
  }
  float area = 0.5f * fabsf(s2);
  return (n >= 3) ? area : 0.f;
}

// ---------------------------------------------------------------------------
// Kernel 2: one wave per 16-row x 32-col tile of the KPRE x KPRE pair matrix.
// WMMA f32 16x16x4 computes pairwise center dot products (K-slots 2,3 zero);
// gate dist^2 <= (r_i + r_j)^2 prunes the exact rotated-IoU evaluation.
// Ballot per accumulator register yields mask bits in exactly the D-matrix
// layout (VGPR r -> rows r / r+8, lanes -> columns).
// ---------------------------------------------------------------------------
__global__ __launch_bounds__(32) void nms_mask_kernel(
    const float* __restrict__ tb, const float* __restrict__ cx,
    const float* __restrict__ cy, const float* __restrict__ pp,
    const float* __restrict__ rad, unsigned* __restrict__ mask) {
  const int b = blockIdx.z;
  const int ti = blockIdx.y;        // row tile (16 rows)
  const int tw = blockIdx.x;        // col word (32 cols = 2 WMMA col tiles)
  const int lane = threadIdx.x;
  const int nl = lane & 15, hf = lane >> 4;

  const float* Cx = cx + (size_t)b * KPRE;
  const float* Cy = cy + (size_t)b * KPRE;
  const float* Pp = pp + (size_t)b * KPRE;
  const float* Rd = rad + (size_t)b * KPRE;
  const float* TB = tb + (size_t)b * KPRE * 7;

  const int c0 = (tw * 2 + 0) * 16 + nl;
  const int c1 = (tw * 2 + 1) * 16 + nl;

  // A (16x4): lanes 0-15 hold K=0,1 for row M=lane; lanes 16-31 (K=2,3) zero.
  v2f av; av[0] = 0.f; av[1] = 0.f;
  if (!hf) { int r = ti * 16 + nl; av[0] = Cx[r]; av[1] = Cy[r]; }
  // B (4x16) per col tile: lanes 0-15 K=0,1 for col N=lane; K=2,3 zero.
  v2f bv0; bv0[0] = 0.f; bv0[1] = 0.f;
  v2f bv1; bv1[0] = 0.f; bv1[1] = 0.f;
  if (!hf) {
    bv0[0] = Cx[c0]; bv0[1] = Cy[c0];
    bv1[0] = Cx[c1]; bv1[1] = Cy[c1];
  }
  v8f acc = {};
  v8f d0 = __builtin_amdgcn_wmma_f32_16x16x4_f32(false, av, false, bv0,
                                                 (short)0, acc, false, false);
  v8f d1 = __builtin_amdgcn_wmma_f32_16x16x4_f32(false, av, false, bv1,
                                                 (short)0, acc, false, false);

  // Per-lane column boxes (same column for both wave halves): build once.
  float colb0[7], colb1[7];
#pragma unroll
  for (int q = 0; q < 7; ++q) { colb0[q] = TB[c0 * 7 + q]; colb1[q] = TB[c1 * 7 + q]; }
  RBox CB0, CB1;
  mk_rbox(colb0, CB0);
  mk_rbox(colb1, CB1);
  const float pc0 = Pp[c0], pc1 = Pp[c1];
  const float rc0 = Rd[c0], rc1 = Rd[c1];

#pragma unroll
  for (int r = 0; r < 8; ++r) {
    const int row = ti * 16 + r + hf * 8;
    float rowb[7];
#pragma unroll
    for (int q = 0; q < 7; ++q) rowb[q] = TB[row * 7 + q];
    RBox RB;
    mk_rbox(rowb, RB);
    const float pr = Pp[row], rr = Rd[row];

    bool p0 = false, p1 = false;
    {
      float dist2 = pr + pc0 - 2.f * d0[r];
      float rs = rr + rc0 + 1e-3f;
      if (dist2 <= rs * rs) {
        float inter = rotated_inter_area(RB, CB0);
        float uni = fmaxf(RB.area + CB0.area - inter, EPSF);
        p0 = (inter / uni) > NMS_TH;
      }
    }
    {
      float dist2 = pr + pc1 - 2.f * d1[r];
      float rs = rr + rc1 + 1e-3f;
      if (dist2 <= rs * rs) {
        float inter = rotated_inter_area(RB, CB1);
        float uni = fmaxf(RB.area + CB1.area - inter, EPSF);
        p1 = (inter / uni) > NMS_TH;
      }
    }
    unsigned bal0 = (unsigned)__ballot((int)p0);  // bits 0-15: row r, 16-31: row r+8
    unsigned bal1 = (unsigned)__ballot((int)p1);
    // rows of this word: low halves -> row ti*16+r, high halves -> row ti*16+8+r
    if (lane == r) {
      unsigned w = (bal0 & 0xFFFFu) | (bal1 << 16);
      mask[((size_t)b * KPRE + ti * 16 + r) * 32 + tw] = w;
    }
    if (lane == 16 + r) {
      unsigned w = (bal0 >> 16) | (bal1 & 0xFFFF0000u);
      mask[((size_t)b * KPRE + ti * 16 + 8 + r) * 32 + tw] = w;
    }
  }
}

// ---------------------------------------------------------------------------
// Kernel 3: per-batch sequential greedy suppression over the bitmask (one
// wave32, lane w owns mask word w), prefix-sum compaction of first POST
// survivors, zero/one-init of outputs (labels background = 0+1 = 1).
// ---------------------------------------------------------------------------
__global__ __launch_bounds__(1024) void nms_scan_kernel(
    const unsigned* __restrict__ mask, const float* __restrict__ tb,
    const float* __restrict__ ts, const int* __restrict__ tl,
    float* __restrict__ out) {
  const int b = blockIdx.x;
  const int tid = threadIdx.x;
  __shared__ unsigned keepw[32];
  __shared__ unsigned pref[33];

  float* rois = out;                                  // [B][POST][7]
  float* rscores = out + (size_t)BB * POST * 7;       // [B][POST]
  float* rlabels = rscores + (size_t)BB * POST;       // [B][POST] (as float)

  for (int i = tid; i < POST * 7; i += 1024) rois[(size_t)b * POST * 7 + i] = 0.f;
  if (tid < POST) {
    rscores[(size_t)b * POST + tid] = 0.f;
    rlabels[(size_t)b * POST + tid] = 1.f;   // reference: zeros + 1
  }
  __syncthreads();

  if (tid < 32) {
    unsigned rem = 0u;                       // suppressed bits in my word
    const unsigned* mb = mask + (size_t)b * KPRE * 32;
    for (int i = 0; i < KPRE; ++i) {
      int wi = i >> 5, bi = i & 31;
      unsigned rw = (unsigned)__shfl((int)rem, wi, 32);
      if (!((rw >> bi) & 1u)) {              // row i still kept
        unsigned m = mb[(size_t)i * 32 + tid];
        if (tid < wi) m = 0u;
        else if (tid == wi) m &= ~((2u << bi) - 1u);  // only j > i
        rem |= m;
      }
    }
    keepw[tid] = ~rem;
  }
  __syncthreads();
  if (tid == 0) {
    unsigned a = 0u;
    for (int w = 0; w < 32; ++w) { pref[w] = a; a += (unsigned)__popc(keepw[w]); }
    pref[32] = a;
  }
  __syncthreads();

  // tid = candidate index 0..KPRE-1
  int w = tid >> 5, bi = tid & 31;
  if ((keepw[w] >> bi) & 1u) {
    unsigned rank = pref[w] + (unsigned)__popc(keepw[w] & ((1u << bi) - 1u));
    if (rank < POST) {
#pragma unroll
      for (int q = 0; q < 7; ++q)
        rois[((size_t)b * POST + rank) * 7 + q] = tb[((size_t)b * KPRE + tid) * 7 + q];
      rscores[(size_t)b * POST + rank] = ts[(size_t)b * KPRE + tid];
      rlabels[(size_t)b * POST + rank] = (float)(tl[(size_t)b * KPRE + tid] + 1);
    }
  }
}

// ---------------------------------------------------------------------------
extern "C" void kernel_launch(void* const* d_in, const int* in_sizes, int n_in,
                              void* d_out, int out_size, void* d_ws, size_t ws_size,
                              hipStream_t stream) {
  const float* boxes = (const float*)d_in[0];  // (B,N,7)
  const float* cls = (const float*)d_in[1];    // (B,N,C)
  float* out = (float*)d_out;

  char* ws = (char*)d_ws;
  size_t off = 0;
  unsigned long long* keys = (unsigned long long*)(ws + off); off += (size_t)BB * NN * 8;
  float* tb = (float*)(ws + off); off += (size_t)BB * KPRE * 7 * 4;
  float* ts = (float*)(ws + off); off += (size_t)BB * KPRE * 4;
  int* tl = (int*)(ws + off); off += (size_t)BB * KPRE * 4;
  float* cxa = (float*)(ws + off); off += (size_t)BB * KPRE * 4;
  float* cya = (float*)(ws + off); off += (size_t)BB * KPRE * 4;
  float* ppa = (float*)(ws + off); off += (size_t)BB * KPRE * 4;
  float* rda = (float*)(ws + off); off += (size_t)BB * KPRE * 4;
  unsigned* mask = (unsigned*)(ws + off); off += (size_t)BB * KPRE * 32 * 4;

  score_key_kernel<<<dim3(NN / 256, BB), 256, 0, stream>>>(cls, keys);

  for (int k = 2; k <= NN; k <<= 1)
    for (int j = k >> 1; j > 0; j >>= 1)
      bitonic_step<<<dim3(NN / 256, BB), 256, 0, stream>>>(keys, k, j);

  gather_kernel<<<dim3(KPRE / 256, BB), 256, 0, stream>>>(
      boxes, cls, keys, tb, ts, tl, cxa, cya, ppa, rda);

  nms_mask_kernel<<<dim3(KPRE / 32, KPRE / 16, BB), 32, 0, stream>>>(
      tb, cxa, cya, ppa, rda, mask);

  nms_scan_kernel<<<BB, 1024, 0, stream>>>(mask, tb, ts, tl, out);
}